// CrossLayerLight_81381040324822
// MI455X (gfx1250) — compile-verified
//
#include <hip/hip_runtime.h>
#include <hip/hip_bf16.h>
#include <stddef.h>

// ---------------------------------------------------------------------------
// CrossLayerLight for MI455X (gfx1250, wave32).
//  K1: brute-force KNN (K=16), LDS point tiles + register insertion sort,
//      global_prefetch of the next tile overlapped with compute.
//  K2: fused gather + pos-encode + 2x (16x64 @ 64x64) WMMA MLP + max-over-K.
//      One wave owns one (dir,b,n) group (straight-line body: no LICM spills);
//      X tile in LDS (stride 68), weights pre-packed in B-fragment order so
//      each WMMA B operand is one aligned ds_load_b64. 64 wmma per layer.
//  K3: [16 rows x 64] @ [64 x 128] WMMA projection + bias -> d_out.
// ---------------------------------------------------------------------------

typedef float v2f __attribute__((ext_vector_type(2)));
typedef float v8f __attribute__((ext_vector_type(8)));

#define NPTS   8192
#define BATCH  2
#define KNN    16
#define CH     64
#define XSTRIDE 68    // padded row stride for 16x64 LDS tiles (bank-conflict free)
#define WPSTRIDE 130  // pair-packed 64x64 W: row p=k/2 holds (k,k+1) x 64 outs
#define TPSTRIDE 258  // pair-packed 64x128 W: row p=k/2 holds (k,k+1) x 128 outs

// ---------------------------------------------------------------------------
// Kernel 1: KNN.  grid = (NPTS/256, BATCH, 2 dirs), block = 256.
// ---------------------------------------------------------------------------
__global__ void knn_kernel(const float* __restrict__ pc1,
                           const float* __restrict__ pc2,
                           int* __restrict__ idxbuf)
{
    __shared__ float sx[256], sy[256], sz[256];
    const int dir = blockIdx.z;
    const int b   = blockIdx.y;
    const float* qp = dir ? pc2 : pc1;
    const float* pp = dir ? pc1 : pc2;

    const int q = blockIdx.x * 256 + threadIdx.x;
    const float* qv = qp + ((size_t)b * NPTS + q) * 3;
    const float qx = qv[0], qy = qv[1], qz = qv[2];

    float bd[KNN];
    int   bi[KNN];
#pragma unroll
    for (int s = 0; s < KNN; ++s) { bd[s] = 3.4e38f; bi[s] = 0; }

    for (int m0 = 0; m0 < NPTS; m0 += 256) {
        __syncthreads();
        const float* pv = pp + ((size_t)b * NPTS + m0 + threadIdx.x) * 3;
        sx[threadIdx.x] = pv[0];
        sy[threadIdx.x] = pv[1];
        sz[threadIdx.x] = pv[2];
        __syncthreads();
        if (m0 + 256 < NPTS) {   // overlap next tile's HBM fetch with compute
            __builtin_prefetch(pp + ((size_t)b * NPTS + m0 + 256 + threadIdx.x) * 3, 0, 0);
        }
        for (int j = 0; j < 256; ++j) {
            const float dx = sx[j] - qx, dy = sy[j] - qy, dz = sz[j] - qz;
            const float d = dx * dx + dy * dy + dz * dz;
            if (d < bd[KNN - 1]) {
                bd[KNN - 1] = d; bi[KNN - 1] = m0 + j;
#pragma unroll
                for (int s = KNN - 1; s >= 1; --s) {
                    if (bd[s] < bd[s - 1]) {
                        const float td = bd[s]; bd[s] = bd[s - 1]; bd[s - 1] = td;
                        const int   ti = bi[s]; bi[s] = bi[s - 1]; bi[s - 1] = ti;
                    }
                }
            }
        }
    }
    int* ob = idxbuf + (((size_t)(dir * BATCH + b) * NPTS + q) * KNN);
#pragma unroll
    for (int s = 0; s < KNN; ++s) ob[s] = bi[s];
}

// ---------------------------------------------------------------------------
// Wave-level in-place 16x64 @ 64x64 layer: X = leaky(X @ W^T + b).
// A frag (16x4 f32): lane L holds row L%16; VGPR v -> K = v + 2*(L>>4).
// B frag from pair-packed WP: WP[p*WPSTRIDE + 2*o + e] = w[o][2p+e];
//   chunk c, half h -> p = 2c + h, one aligned ds_load_b64 per fragment.
// C/D (16x16 f32):   lane L holds col L%16; VGPR v -> row v + 8*(L>>4).
// ---------------------------------------------------------------------------
__device__ __forceinline__ void mlp_layer_16x64(float* X, const float* WP,
                                                const float* bias, int lane)
{
    const int row  = lane & 15;
    const int half = lane >> 4;
    v8f acc[4] = {v8f{}, v8f{}, v8f{}, v8f{}};
    const float* arow = X + row * XSTRIDE + 2 * half;
    const float* wrow = WP + half * WPSTRIDE + 2 * row;
#pragma unroll
    for (int c = 0; c < 16; ++c) {
        const v2f a = *(const v2f*)(arow + 4 * c);   // X[row][4c + 2h + {0,1}]
#pragma unroll
        for (int t = 0; t < 4; ++t) {
            const v2f bf = *(const v2f*)(wrow + (2 * c) * WPSTRIDE + 32 * t);
            acc[t] = __builtin_amdgcn_wmma_f32_16x16x4_f32(
                false, a, false, bf, (short)0, acc[t], false, false);
        }
    }
#pragma unroll
    for (int t = 0; t < 4; ++t) {
        const int o = 16 * t + row;
        const float bv = bias[o];
#pragma unroll
        for (int v = 0; v < 8; ++v) {
            const int m = v + 8 * half;
            float y = acc[t][v] + bv;
            y = (y >= 0.f) ? y : 0.1f * y;
            X[m * XSTRIDE + o] = y;
        }
    }
}

// ---------------------------------------------------------------------------
// Kernel 2: gather + pos-encode + MLP(64->64->64) + max over K.
// grid = 8192 blocks, block = 128 (4 waves); one wave per (dir,b,n) group.
// ---------------------------------------------------------------------------
__global__ void group_mlp_kernel(const float* __restrict__ pc1,
                                 const float* __restrict__ pc2,
                                 const float* __restrict__ feat1,
                                 const float* __restrict__ feat2,
                                 const float* __restrict__ pos_w,
                                 const float* __restrict__ pos_b,
                                 const float* __restrict__ w0g,
                                 const float* __restrict__ b0g,
                                 const float* __restrict__ w1g,
                                 const float* __restrict__ b1g,
                                 const int* __restrict__ idxbuf,
                                 float* __restrict__ mid)
{
    __shared__ float w0P[32 * WPSTRIDE];
    __shared__ float w1P[32 * WPSTRIDE];
    __shared__ float poswS[64 * 3];
    __shared__ float posbS[64], b0S[64], b1S[64];
    __shared__ float Xbuf[4][16 * XSTRIDE];

    for (int i = threadIdx.x; i < 64 * 64; i += 128) {
        const int o = i >> 6, c = i & 63;           // w[o][c]
        const int p = c >> 1, e = c & 1;            // -> pair-packed frag layout
        w0P[p * WPSTRIDE + 2 * o + e] = w0g[i];
        w1P[p * WPSTRIDE + 2 * o + e] = w1g[i];
    }
    if (threadIdx.x < 64 * 3) poswS[threadIdx.x] = pos_w[threadIdx.x];
    if (threadIdx.x < 64) {
        posbS[threadIdx.x] = pos_b[threadIdx.x];
        b0S[threadIdx.x]   = b0g[threadIdx.x];
        b1S[threadIdx.x]   = b1g[threadIdx.x];
    }
    __syncthreads();

    const int wid  = threadIdx.x >> 5;
    const int lane = threadIdx.x & 31;
    const int g    = blockIdx.x * 4 + wid;          // [0, 32768)
    const int dir  = g >> 14;
    const int rem  = g & 16383;
    const int b    = rem >> 13;
    const int n    = rem & 8191;

    const float* qf = dir ? feat2 : feat1;
    const float* pf = dir ? feat1 : feat2;
    const float* qp = dir ? pc2 : pc1;
    const float* pp = dir ? pc1 : pc2;

    float* X = Xbuf[wid];
    const int k = lane & 15, half = lane >> 4;

    const int ii = idxbuf[(size_t)g * KNN + k];
    const float* qxyz = qp + ((size_t)(b * NPTS) + n) * 3;
    const float* nxyz = pp + ((size_t)(b * NPTS) + ii) * 3;
    const float dx = nxyz[0] - qxyz[0];
    const float dy = nxyz[1] - qxyz[1];
    const float dz = nxyz[2] - qxyz[2];

    const float4* p1v = (const float4*)(qf + ((size_t)(b * NPTS) + n) * CH + half * 32);
    const float4* g2v = (const float4*)(pf + ((size_t)(b * NPTS) + ii) * CH + half * 32);
    float* xrow = X + k * XSTRIDE + half * 32;
#pragma unroll
    for (int c4 = 0; c4 < 8; ++c4) {
        const float4 a  = p1v[c4];
        const float4 gg = g2v[c4];
#pragma unroll
        for (int e = 0; e < 4; ++e) {
            const int c = half * 32 + c4 * 4 + e;
            const float pe = poswS[c * 3 + 0] * dx + poswS[c * 3 + 1] * dy +
                             poswS[c * 3 + 2] * dz + posbS[c];
            float x = (&a.x)[e] + (&gg.x)[e] + pe;
            x = (x >= 0.f) ? x : 0.1f * x;
            xrow[c4 * 4 + e] = x;
        }
    }
    // Tile is private to this wave; DS ops are in-order per wave -> no barrier.
    mlp_layer_16x64(X, w0P, b0S, lane);
    mlp_layer_16x64(X, w1P, b1S, lane);

    // max over the 16 neighbor rows; each lane reduces channels lane, lane+32
#pragma unroll
    for (int cc = 0; cc < 2; ++cc) {
        const int c = lane + 32 * cc;
        float m = X[c];
#pragma unroll
        for (int r = 1; r < 16; ++r) m = fmaxf(m, X[r * XSTRIDE + c]);
        mid[(size_t)g * CH + c] = m;
    }
}

// ---------------------------------------------------------------------------
// Kernel 3: projection [16384 x 64] @ [64 x 128] + bias, per direction.
// grid = (256, 2 dirs), block = 128 (4 waves); one wave per 16-row tile.
// ---------------------------------------------------------------------------
__global__ void proj_kernel(const float* __restrict__ mid,
                            const float* __restrict__ t1w,
                            const float* __restrict__ t1b,
                            const float* __restrict__ t2w,
                            const float* __restrict__ t2b,
                            float* __restrict__ out)
{
    __shared__ float tP[32 * TPSTRIDE];
    __shared__ float tbS[128];
    __shared__ float Xs[4][16 * XSTRIDE];

    const int dir = blockIdx.y;
    const float* tw = dir ? t2w : t1w;
    const float* tb = dir ? t2b : t1b;

    for (int i = threadIdx.x; i < 128 * 64; i += 128) {
        const int o = i >> 6, c = i & 63;           // t[o][c]
        const int p = c >> 1, e = c & 1;
        tP[p * TPSTRIDE + 2 * o + e] = tw[i];
    }
    if (threadIdx.x < 128) tbS[threadIdx.x] = tb[threadIdx.x];
    __syncthreads();

    const int wid  = threadIdx.x >> 5;
    const int lane = threadIdx.x & 31;
    const int row  = lane & 15;
    const int half = lane >> 4;
    const int rbase = (blockIdx.x * 4 + wid) * 16;   // row within this dir
    float* X = Xs[wid];

    const float* src = mid + ((size_t)dir * (BATCH * NPTS) + rbase) * CH;
    {
        const float4* sv = (const float4*)(src + (size_t)row * CH + half * 32);
        float* dst = X + row * XSTRIDE + half * 32;
#pragma unroll
        for (int c4 = 0; c4 < 8; ++c4) *(float4*)(dst + c4 * 4) = sv[c4];
    }

    float* outp = out + ((size_t)dir * (BATCH * NPTS) + rbase) * 128;
    const float* arow = X + row * XSTRIDE + 2 * half;
    const float* wrow = tP + half * TPSTRIDE + 2 * row;
#pragma unroll
    for (int t = 0; t < 8; ++t) {
        v8f acc = v8f{};
#pragma unroll
        for (int c = 0; c < 16; ++c) {
            const v2f a  = *(const v2f*)(arow + 4 * c);
            const v2f bf = *(const v2f*)(wrow + (2 * c) * TPSTRIDE + 32 * t);
            acc = __builtin_amdgcn_wmma_f32_16x16x4_f32(
                false, a, false, bf, (short)0, acc, false, false);
        }
        const int o = 16 * t + row;
        const float bv = tbS[o];
#pragma unroll
        for (int v = 0; v < 8; ++v) {
            const int m = v + 8 * half;
            outp[(size_t)m * 128 + o] = acc[v] + bv;
        }
    }
}

// ---------------------------------------------------------------------------
// Host launcher.
// ---------------------------------------------------------------------------
extern "C" void kernel_launch(void* const* d_in, const int* in_sizes, int n_in,
                              void* d_out, int out_size, void* d_ws, size_t ws_size,
                              hipStream_t stream)
{
    const float* pc1   = (const float*)d_in[0];
    const float* pc2   = (const float*)d_in[1];
    const float* feat1 = (const float*)d_in[2];
    const float* feat2 = (const float*)d_in[3];
    const float* pos_w = (const float*)d_in[4];
    const float* pos_b = (const float*)d_in[5];
    const float* w0    = (const float*)d_in[6];
    const float* b0    = (const float*)d_in[7];
    const float* w1    = (const float*)d_in[8];
    const float* b1    = (const float*)d_in[9];
    const float* t1w   = (const float*)d_in[10];
    const float* t1b   = (const float*)d_in[11];
    const float* t2w   = (const float*)d_in[12];
    const float* t2b   = (const float*)d_in[13];

    // workspace: [0, 2MB) knn indices (2*2*8192*16 int), [2MB, 10MB) maxed feats
    int*   idxbuf = (int*)d_ws;
    float* mid    = (float*)((char*)d_ws + (size_t)2 * BATCH * NPTS * KNN * sizeof(int));

    knn_kernel<<<dim3(NPTS / 256, BATCH, 2), 256, 0, stream>>>(pc1, pc2, idxbuf);

    group_mlp_kernel<<<(2 * BATCH * NPTS) / 4, 128, 0, stream>>>(
        pc1, pc2, feat1, feat2, pos_w, pos_b, w0, b0, w1, b1, idxbuf, mid);

    proj_kernel<<<dim3((BATCH * NPTS) / (4 * 16), 2), 128, 0, stream>>>(
        mid, t1w, t1b, t2w, t2b, (float*)d_out);
}